// STAN_57518202028568
// MI455X (gfx1250) — compile-verified
//
#include <hip/hip_runtime.h>
#include <hip/hip_bf16.h>
#include <math.h>

#define NUM_LOC 20000
#define T_STEPS 24
#define IN_DIM  18
#define H       32
#define PW      15

typedef float v2f __attribute__((ext_vector_type(2)));
typedef float v8f __attribute__((ext_vector_type(8)));

// ---------- ordered-uint transform for float atomic max (handles negatives) ----------
__device__ __forceinline__ unsigned f2key(float f) {
    unsigned u = __float_as_uint(f);
    return (u & 0x80000000u) ? ~u : (u | 0x80000000u);
}
__device__ __forceinline__ float key2f(unsigned k) {
    return __uint_as_float((k & 0x80000000u) ? (k ^ 0x80000000u) : ~k);
}
__device__ __forceinline__ float sigmf(float x) { return 1.0f / (1.0f + expf(-x)); }

// ---------- WMMA node-linear: Z[20000x32] = X[20000xK] @ W^T + b ----------
// One wave computes a 16-row x 32-col tile via V_WMMA_F32_16X16X4_F32.
// A 16x4 f32 layout: lane l(0..15) VGPR0/1 = A[l][k0+0..1]; lanes 16..31 hold k0+2..3.
// B 4x16 = W^T, and since W is (N,K) row-major, loading W rows like A rows gives B.
// K is compile-time: full unroll, unconditional b64 loads; tail (K=18) uses clamped
// addresses + branchless selects so EXEC is never perturbed around the WMMAs.
template <int K>
__global__ void k_linear_wmma(const float* __restrict__ X, int ldx,
                              const float* __restrict__ W,
                              const float* __restrict__ bias,
                              float* __restrict__ Z) {
    int wave = threadIdx.x >> 5;
    int tile = blockIdx.x * (blockDim.x >> 5) + wave;
    if (tile >= NUM_LOC / 16) return;            // whole wave exits together
    int lane = threadIdx.x & 31;
    int half = lane >> 4;
    int l    = lane & 15;

    const float* xr = X + (size_t)(tile * 16 + l) * ldx;
    const float* w0 = W + (size_t)l * K;
    const float* w1 = W + (size_t)(16 + l) * K;

    float bl0 = bias[l], bl1 = bias[16 + l];
    v8f c0, c1;
#pragma unroll
    for (int r = 0; r < 8; ++r) { c0[r] = bl0; c1[r] = bl1; }

#pragma unroll
    for (int k0 = 0; k0 < K; k0 += 4) {
        int ka = k0 + 2 * half;                  // this lane's K pair
        v2f a, b0, b1;
        if (k0 + 4 <= K) {
            // fully in-bounds for every lane: plain 8-byte loads
            a  = *(const v2f*)(xr + ka);
            b0 = *(const v2f*)(w0 + ka);
            b1 = *(const v2f*)(w1 + ka);
        } else {
            // tail chunk (K=18 -> k0=16): clamp address in-bounds, zero by select
            int kc = (ka + 1 < K) ? ka : (K - 2);   // K even; kc is even & in-bounds
            v2f av = *(const v2f*)(xr + kc);
            v2f bv0 = *(const v2f*)(w0 + kc);
            v2f bv1 = *(const v2f*)(w1 + kc);
            bool ok = (ka + 1 < K);                 // whole pair valid or whole pair dead
            a.x  = ok ? av.x  : 0.0f;  a.y  = ok ? av.y  : 0.0f;
            b0.x = ok ? bv0.x : 0.0f;  b0.y = ok ? bv0.y : 0.0f;
            b1.x = ok ? bv1.x : 0.0f;  b1.y = ok ? bv1.y : 0.0f;
        }
        c0 = __builtin_amdgcn_wmma_f32_16x16x4_f32(false, a, false, b0, (short)0, c0, false, false);
        c1 = __builtin_amdgcn_wmma_f32_16x16x4_f32(false, a, false, b1, (short)0, c1, false, false);
    }

    // C/D layout: VGPR r -> row (tile*16 + r + half*8), col = (lane&15) [+16 for c1]
    int orow = tile * 16 + half * 8;
#pragma unroll
    for (int r = 0; r < 8; ++r) {
        Z[(size_t)(orow + r) * H + l]      = c0[r];
        Z[(size_t)(orow + r) * H + 16 + l] = c1[r];
    }
}

// ---------- per-node attention scalars + per-timestep-layer buffer init ----------
__global__ void k_att_init(const float* __restrict__ Z,
                           const float* __restrict__ aW, const float* __restrict__ ab,
                           float* __restrict__ es, float* __restrict__ ed,
                           unsigned* __restrict__ mkey, float* __restrict__ den,
                           float* __restrict__ agg, unsigned* __restrict__ poolkey,
                           int do_pool) {
    int n = blockIdx.x * blockDim.x + threadIdx.x;
    if (n >= NUM_LOC) return;
    const float* zr = Z + (size_t)n * H;
    float s1 = 0.0f, s2 = 0.0f;
#pragma unroll
    for (int j = 0; j < H; ++j) {
        float v = zr[j];
        s1 += v * aW[j];
        s2 += v * aW[H + j];
    }
    es[n] = s1;
    ed[n] = s2 + ab[0];
    mkey[n] = 0u;          // smallest ordered key
    den[n]  = 0.0f;
    float* ar = agg + (size_t)n * H;
#pragma unroll
    for (int j = 0; j < H; ++j) ar[j] = 0.0f;
    if (do_pool && n < H) poolkey[n] = 0u;
}

// ---------- edge pass 1: segment max of leaky_relu(es[src]+ed[dst]) ----------
__global__ void k_edge_max(const int* __restrict__ src, const int* __restrict__ dst,
                           const float* __restrict__ es, const float* __restrict__ ed,
                           unsigned* __restrict__ mkey, int E) {
    int e = blockIdx.x * blockDim.x + threadIdx.x;
    if (e >= E) return;
    float x = es[src[e]] + ed[dst[e]];
    float lr = (x > 0.0f) ? x : 0.01f * x;       // jax leaky_relu default slope
    atomicMax(&mkey[dst[e]], f2key(lr));
}

// ---------- edge pass 2: ex = exp(e - m[dst]); den[dst] += ex ----------
__global__ void k_edge_expsum(const int* __restrict__ src, const int* __restrict__ dst,
                              const float* __restrict__ es, const float* __restrict__ ed,
                              const unsigned* __restrict__ mkey, float* __restrict__ den,
                              float* __restrict__ exb, int E) {
    int e = blockIdx.x * blockDim.x + threadIdx.x;
    if (e >= E) return;
    int d = dst[e];
    float x = es[src[e]] + ed[d];
    float lr = (x > 0.0f) ? x : 0.01f * x;
    float ex = expf(lr - key2f(mkey[d]));
    exb[e] = ex;
    atomicAdd(&den[d], ex);
}

// ---------- edge pass 3: agg[dst][j] += (ex/den[dst]) * z[src][j]; 32 lanes per edge ----------
__global__ void k_edge_agg(const int* __restrict__ src, const int* __restrict__ dst,
                           const float* __restrict__ Z, const float* __restrict__ exb,
                           const float* __restrict__ den, float* __restrict__ agg, int E) {
    int e = blockIdx.x * (blockDim.x >> 5) + (threadIdx.x >> 5);
    if (e >= E) return;
    int j = threadIdx.x & 31;
    int s = src[e], d = dst[e];
    float alpha = exb[e] / den[d];
    atomicAdd(&agg[(size_t)d * H + j], alpha * Z[(size_t)s * H + j]);
}

// ---------- elu (+ optional column max-pool into ordered keys) ----------
__global__ void k_elu(const float* __restrict__ agg, float* __restrict__ c,
                      unsigned* __restrict__ poolkey) {
    int i = blockIdx.x * blockDim.x + threadIdx.x;
    if (i >= NUM_LOC * H) return;
    float v = agg[i];
    float o = (v > 0.0f) ? v : (expf(v) - 1.0f);
    c[i] = o;
    if (poolkey) atomicMax(&poolkey[i & (H - 1)], f2key(o));
}

// ---------- GRU cell + prediction heads (single 64-thread block) ----------
__global__ void k_gru_head(const unsigned* __restrict__ poolkey,
                           const float* __restrict__ h0, float* __restrict__ h_ws,
                           const float* __restrict__ Wih, const float* __restrict__ Whh,
                           const float* __restrict__ bih, const float* __restrict__ bhh,
                           const float* __restrict__ WI, const float* __restrict__ bI,
                           const float* __restrict__ WR, const float* __restrict__ bR,
                           const float* __restrict__ Wsm, const float* __restrict__ bsm,
                           const float* __restrict__ cI, const float* __restrict__ cR,
                           float* __restrict__ out, float* __restrict__ ab_ws, int t) {
    __shared__ float pooled[H], hold[H], hnew[H];
    int j = threadIdx.x;
    if (j < H) {
        pooled[j] = key2f(poolkey[j]);
        hold[j]   = (t == 0) ? h0[j] : h_ws[j];
    }
    __syncthreads();
    if (j < H) {
        float gir = bih[j], giz = bih[H + j], gin = bih[2 * H + j];
        float ghr = bhh[j], ghz = bhh[H + j], ghn = bhh[2 * H + j];
#pragma unroll
        for (int k = 0; k < H; ++k) {
            float p = pooled[k], hh = hold[k];
            gir += Wih[(size_t)j * H + k] * p;
            giz += Wih[(size_t)(H + j) * H + k] * p;
            gin += Wih[(size_t)(2 * H + j) * H + k] * p;
            ghr += Whh[(size_t)j * H + k] * hh;
            ghz += Whh[(size_t)(H + j) * H + k] * hh;
            ghn += Whh[(size_t)(2 * H + j) * H + k] * hh;
        }
        float r  = sigmf(gir + ghr);
        float zg = sigmf(giz + ghz);
        float nn = tanhf(gin + r * ghn);
        float hn = (1.0f - zg) * nn + zg * hold[j];
        hnew[j] = hn;
        h_ws[j] = hn;
        if (t == T_STEPS - 1)
            out[(size_t)2 * PW * T_STEPS + (size_t)2 * NUM_LOC * T_STEPS * PW + j] = hn;
    }
    __syncthreads();
    float cIt = cI[t], cRt = cR[t];
    if (j < PW) {                                         // pred_I -> new_I[0][t][j]
        float a = bI[j];
#pragma unroll
        for (int k = 0; k < H; ++k) a += WI[(size_t)j * (H + 2) + k] * hnew[k];
        a += WI[(size_t)j * (H + 2) + H] * cIt + WI[(size_t)j * (H + 2) + H + 1] * cRt;
        out[(size_t)t * PW + j] = a;
    } else if (j < 2 * PW) {                              // pred_R -> new_R[0][t][jj]
        int jj = j - PW;
        float a = bR[jj];
#pragma unroll
        for (int k = 0; k < H; ++k) a += WR[(size_t)jj * (H + 2) + k] * hnew[k];
        a += WR[(size_t)jj * (H + 2) + H] * cIt + WR[(size_t)jj * (H + 2) + H + 1] * cRt;
        out[(size_t)PW * T_STEPS + (size_t)t * PW + jj] = a;
    } else if (j < 2 * PW + 2) {                          // alpha, beta
        int jj = j - 2 * PW;
        float a = bsm[jj];
#pragma unroll
        for (int k = 0; k < H; ++k) a += Wsm[(size_t)jj * (H + 2) + k] * hnew[k];
        a += Wsm[(size_t)jj * (H + 2) + H] * cIt + Wsm[(size_t)jj * (H + 2) + H + 1] * cRt;
        ab_ws[jj] = sigmf(a);
    }
}

// ---------- per-node SIR rollout: writes phy_I[n][t][:], phy_R[n][t][:] ----------
__global__ void k_sir(const float* __restrict__ Nin, const float* __restrict__ I,
                      const float* __restrict__ R, const float* __restrict__ ab_ws,
                      float* __restrict__ out, int t) {
    int n = blockIdx.x * blockDim.x + threadIdx.x;
    if (n >= NUM_LOC) return;
    float alpha = ab_ws[0], beta = ab_ws[1];
    float Nv = Nin[n];
    float lI = I[t], lR = R[t];
    const size_t PHYI = (size_t)2 * PW * T_STEPS;
    const size_t PHYR = PHYI + (size_t)NUM_LOC * T_STEPS * PW;
    float* oI = out + PHYI + (size_t)n * T_STEPS * PW + (size_t)t * PW;
    float* oR = out + PHYR + (size_t)n * T_STEPS * PW + (size_t)t * PW;
#pragma unroll
    for (int s = 0; s < PW; ++s) {
        float lS = Nv - lI - lR;
        float dI = alpha * lI * (lS / Nv) - beta * lI;
        float dR = beta * lI;
        oI[s] = dI;
        oR[s] = dR;
        lI += dI;
        lR += dR;
    }
}

extern "C" void kernel_launch(void* const* d_in, const int* in_sizes, int n_in,
                              void* d_out, int out_size, void* d_ws, size_t ws_size,
                              hipStream_t stream) {
    const float* dynamic = (const float*)d_in[0];
    const float* cI  = (const float*)d_in[1];
    const float* cR  = (const float*)d_in[2];
    const float* Nin = (const float*)d_in[3];
    const float* I   = (const float*)d_in[4];
    const float* R   = (const float*)d_in[5];
    const float* h0  = (const float*)d_in[6];
    const float* W1  = (const float*)d_in[7];
    const float* b1  = (const float*)d_in[8];
    const float* aW1 = (const float*)d_in[9];
    const float* ab1 = (const float*)d_in[10];
    const float* W2  = (const float*)d_in[11];
    const float* b2  = (const float*)d_in[12];
    const float* aW2 = (const float*)d_in[13];
    const float* ab2 = (const float*)d_in[14];
    const float* Wih = (const float*)d_in[15];
    const float* Whh = (const float*)d_in[16];
    const float* bih = (const float*)d_in[17];
    const float* bhh = (const float*)d_in[18];
    const float* WI  = (const float*)d_in[19];
    const float* bI  = (const float*)d_in[20];
    const float* WR  = (const float*)d_in[21];
    const float* bR  = (const float*)d_in[22];
    const float* Wsm = (const float*)d_in[23];
    const float* bsm = (const float*)d_in[24];
    const int* esrc  = (const int*)d_in[25];
    const int* edst  = (const int*)d_in[26];
    const int E = in_sizes[25];
    float* out = (float*)d_out;

    // workspace layout (floats)
    float* ws = (float*)d_ws;
    float*    zbuf = ws;                               // 20000*32
    float*    cbuf = ws + 640000;                      // 20000*32
    float*    agg  = ws + 1280000;                     // 20000*32
    float*    es   = ws + 1920000;                     // 20000
    float*    ed   = ws + 1940000;                     // 20000
    float*    den  = ws + 1960000;                     // 20000
    unsigned* mkey = (unsigned*)(ws + 1980000);        // 20000
    float*    exb  = ws + 2000000;                     // E (<= 660000)
    unsigned* pkey = (unsigned*)(ws + 2660000);        // 32
    float*    h_ws = ws + 2660032;                     // 32
    float*    ab_ws = ws + 2660064;                    // 2

    const int tilesB = (NUM_LOC / 16 + 7) / 8;          // 8 waves/block
    const int nodeB  = (NUM_LOC + 255) / 256;
    const int edgeB  = (E + 255) / 256;
    const int eaggB  = (E + 7) / 8;
    const int eluB   = (NUM_LOC * H + 255) / 256;

    for (int t = 0; t < T_STEPS; ++t) {
        // ---- GAT layer 1 ----
        k_linear_wmma<IN_DIM><<<tilesB, 256, 0, stream>>>(dynamic + (size_t)t * IN_DIM,
                                                          T_STEPS * IN_DIM, W1, b1, zbuf);
        k_att_init<<<nodeB, 256, 0, stream>>>(zbuf, aW1, ab1, es, ed, mkey, den, agg, pkey, 0);
        k_edge_max<<<edgeB, 256, 0, stream>>>(esrc, edst, es, ed, mkey, E);
        k_edge_expsum<<<edgeB, 256, 0, stream>>>(esrc, edst, es, ed, mkey, den, exb, E);
        k_edge_agg<<<eaggB, 256, 0, stream>>>(esrc, edst, zbuf, exb, den, agg, E);
        k_elu<<<eluB, 256, 0, stream>>>(agg, cbuf, nullptr);
        // ---- GAT layer 2 ----
        k_linear_wmma<H><<<tilesB, 256, 0, stream>>>(cbuf, H, W2, b2, zbuf);
        k_att_init<<<nodeB, 256, 0, stream>>>(zbuf, aW2, ab2, es, ed, mkey, den, agg, pkey, 1);
        k_edge_max<<<edgeB, 256, 0, stream>>>(esrc, edst, es, ed, mkey, E);
        k_edge_expsum<<<edgeB, 256, 0, stream>>>(esrc, edst, es, ed, mkey, den, exb, E);
        k_edge_agg<<<eaggB, 256, 0, stream>>>(esrc, edst, zbuf, exb, den, agg, E);
        k_elu<<<eluB, 256, 0, stream>>>(agg, cbuf, pkey);   // + column max-pool
        // ---- GRU + heads + SIR ----
        k_gru_head<<<1, 64, 0, stream>>>(pkey, h0, h_ws, Wih, Whh, bih, bhh,
                                         WI, bI, WR, bR, Wsm, bsm, cI, cR, out, ab_ws, t);
        k_sir<<<nodeB, 256, 0, stream>>>(Nin, I, R, ab_ws, out, t);
    }
}